// SparseTF_88691074662715
// MI455X (gfx1250) — compile-verified
//
#include <hip/hip_runtime.h>

// Problem constants (match reference)
#define INPUT_DIM 20000
#define UNITS     4096
#define NNZ       800000
#define BATCH     1024

// K is densified in chunks so the active W slice (65.5 MB) + x (82 MB) fit in
// the 192 MB L2; also caps workspace at ~82 MB instead of 328 MB.
#define K_CHUNK   4000
#define N_CHUNKS  5            // 5 * 4000 = 20000

// GEMM tiling: 256 threads = 8 wave32, block tile 128x128, K-tile 16.
#define BM 128
#define BN 128
#define BK 16
#define ASTR  18               // padded LDS stride for A tile (128 x 16 f32)
#define BSTRP 144              // padded stride (in float2) for K-pair-interleaved B tile

typedef float v2f __attribute__((ext_vector_type(2)));
typedef float v8f __attribute__((ext_vector_type(8)));

__global__ void zero_ws_kernel(float4* __restrict__ p, int n4) {
  int i = blockIdx.x * blockDim.x + threadIdx.x;
  if (i < n4) p[i] = make_float4(0.f, 0.f, 0.f, 0.f);
}

__global__ void scatter_chunk_kernel(const int* __restrict__ ind,
                                     const float* __restrict__ val,
                                     float* __restrict__ Wc, int kbase) {
  int i = blockIdx.x * blockDim.x + threadIdx.x;
  if (i >= NNZ) return;
  int r = ind[2 * i];
  int c = ind[2 * i + 1];
  unsigned rl = (unsigned)(r - kbase);
  if (rl < (unsigned)K_CHUNK) {
    atomicAdd(Wc + (size_t)rl * UNITS + (size_t)c, val[i]);
  }
}

__global__ __launch_bounds__(256)
void gemm_chunk_kernel(const float* __restrict__ X,
                       const float* __restrict__ Wc,
                       const float* __restrict__ bias,
                       float* __restrict__ accum,
                       float* __restrict__ out,
                       int chunk) {
  __shared__ float As[BM * ASTR];      // A tile: [m][k], padded row stride
  __shared__ v2f   Bp[8 * BSTRP];      // B tile: [k/2][n] -> {B[2k'][n], B[2k'+1][n]}

  const int tid  = threadIdx.x;
  const int lane = tid & 31;
  const int wave = tid >> 5;
  const int wm   = wave >> 2;          // 0..1 : 64-row wave slice
  const int wn   = wave & 3;           // 0..3 : 32-col wave slice
  const int m0   = blockIdx.y * BM;
  const int n0   = blockIdx.x * BN;

  const int mlan  = lane & 15;         // M (A) / N (B) index within 16
  const int khalf = lane >> 4;         // K pair select: {0,1} vs {2,3}

  // global -> LDS staging maps
  const int arow = tid >> 1;           // 0..127 batch row in tile
  const int acol = (tid & 1) * 8;      // 0 or 8
  const int kp   = tid >> 5;           // 0..7   B pair-row
  const int bn   = (tid & 31) * 4;     // 0..124 B col group (4 cols)

  const int kbase = chunk * K_CHUNK;
  const float* xptr = X + (size_t)(m0 + arow) * INPUT_DIM + (size_t)kbase + acol;
  const float* wbase = Wc + (size_t)n0 + bn;

  v8f c[4][2];
  #pragma unroll
  for (int mi = 0; mi < 4; ++mi)
    #pragma unroll
    for (int ni = 0; ni < 2; ++ni)
      #pragma unroll
      for (int j = 0; j < 8; ++j)
        c[mi][ni] [j] = 0.f;

  // ---- software pipeline: prefetch tile kt=0 into registers ----
  float4 a0 = *(const float4*)(xptr + 0);
  float4 a1 = *(const float4*)(xptr + 4);
  const float* wp0 = wbase + (size_t)(2 * kp) * UNITS;
  float4 w0 = *(const float4*)(wp0);
  float4 w1 = *(const float4*)(wp0 + UNITS);

  for (int kt = 0; kt < K_CHUNK; kt += BK) {
    __syncthreads();                   // previous iteration's LDS reads done
    {
      float* ap = &As[arow * ASTR + acol];
      ap[0] = a0.x; ap[1] = a0.y; ap[2] = a0.z; ap[3] = a0.w;
      ap[4] = a1.x; ap[5] = a1.y; ap[6] = a1.z; ap[7] = a1.w;
      // K-pair interleave: 32 contiguous bytes, 16B aligned
      float* bp = (float*)&Bp[kp * BSTRP + bn];
      bp[0] = w0.x; bp[1] = w1.x; bp[2] = w0.y; bp[3] = w1.y;
      bp[4] = w0.z; bp[5] = w1.z; bp[6] = w0.w; bp[7] = w1.w;
    }
    __syncthreads();

    // Issue next tile's global loads; they overlap the 32 WMMAs below.
    const int knext = kt + BK;
    if (knext < K_CHUNK) {             // wave-uniform branch (scalar)
      a0 = *(const float4*)(xptr + knext);
      a1 = *(const float4*)(xptr + knext + 4);
      const float* wq = wbase + (size_t)(knext + 2 * kp) * UNITS;
      w0 = *(const float4*)(wq);
      w1 = *(const float4*)(wq + UNITS);
    }

    #pragma unroll
    for (int kk = 0; kk < 4; ++kk) {
      const int kb  = kk * 4 + khalf * 2;  // A: K base for this lane half
      const int kpf = kk * 2 + khalf;      // B: pair-row for this lane half
      v2f a[4];
      v2f b[2];
      // A 16x4 f32 layout: lanes 0-15 -> K={kb,kb+1}, lanes 16-31 -> K={kb+2,kb+3}
      #pragma unroll
      for (int mi = 0; mi < 4; ++mi) {
        const float* s = &As[(wm * 64 + mi * 16 + mlan) * ASTR + kb];
        a[mi].x = s[0];
        a[mi].y = s[1];
      }
      // B fragment = one b64 load thanks to pair-interleaved layout
      #pragma unroll
      for (int ni = 0; ni < 2; ++ni) {
        b[ni] = Bp[kpf * BSTRP + wn * 32 + ni * 16 + mlan];
      }
      #pragma unroll
      for (int mi = 0; mi < 4; ++mi)
        #pragma unroll
        for (int ni = 0; ni < 2; ++ni)
          c[mi][ni] = __builtin_amdgcn_wmma_f32_16x16x4_f32(
              false, a[mi], false, b[ni], (short)0, c[mi][ni], false, false);
    }
  }

  // Epilogue: C/D layout -> VGPR j holds M=j (lanes 0-15) / M=j+8 (lanes 16-31), N=lane&15
  const bool first = (chunk == 0);
  const bool last  = (chunk == N_CHUNKS - 1);
  #pragma unroll
  for (int mi = 0; mi < 4; ++mi) {
    #pragma unroll
    for (int ni = 0; ni < 2; ++ni) {
      const int col = n0 + wn * 32 + ni * 16 + mlan;
      const int rb  = m0 + wm * 64 + mi * 16 + khalf * 8;
      #pragma unroll
      for (int j = 0; j < 8; ++j) {
        const size_t idx = (size_t)(rb + j) * UNITS + (size_t)col;
        float v = c[mi][ni][j];
        if (!first) v += accum[idx];
        if (last) {
          out[idx] = tanhf(v + bias[col]);
        } else {
          accum[idx] = v;
        }
      }
    }
  }
}

extern "C" void kernel_launch(void* const* d_in, const int* in_sizes, int n_in,
                              void* d_out, int out_size, void* d_ws, size_t ws_size,
                              hipStream_t stream) {
  const float* x    = (const float*)d_in[0];   // [1024, 20000] f32
  const float* kv   = (const float*)d_in[1];   // [800000] f32
  const float* bias = (const float*)d_in[2];   // [4096] f32
  const int*   ind  = (const int*)d_in[3];     // [800000, 2] i32 (row, col)
  float* out = (float*)d_out;                  // [1024, 4096] f32

  // Workspace layout: W chunk (65.5 MB) then f32 accumulator (16.8 MB)
  float* Wc    = (float*)d_ws;
  float* accum = Wc + (size_t)K_CHUNK * UNITS;

  const int wc4 = (K_CHUNK * UNITS) / 4;       // float4 count for zeroing
  dim3 zgrid((wc4 + 255) / 256);
  dim3 sgrid((NNZ + 255) / 256);
  dim3 ggrid(UNITS / BN, BATCH / BM);          // 32 x 8 = 256 blocks

  for (int chunk = 0; chunk < N_CHUNKS; ++chunk) {
    zero_ws_kernel<<<zgrid, 256, 0, stream>>>((float4*)Wc, wc4);
    scatter_chunk_kernel<<<sgrid, 256, 0, stream>>>(ind, kv, Wc, chunk * K_CHUNK);
    gemm_chunk_kernel<<<ggrid, 256, 0, stream>>>(x, Wc, bias, accum, out, chunk);
  }
  (void)in_sizes; (void)n_in; (void)out_size; (void)ws_size;
}